// LLFullObjectCondensation_35768487641759
// MI455X (gfx1250) — compile-verified
//
#include <hip/hip_runtime.h>
#include <hip/hip_bf16.h>
#include <math.h>

typedef __attribute__((ext_vector_type(2))) float v2f;
typedef __attribute__((ext_vector_type(8))) float v8f;

#define KOBJ 256
#define BETA_CLIP_C (1.0f - 1e-5f)
#define Q_MIN_C 0.5f

// ---- workspace layout (bytes) ----
#define OFF_AMAX    0        // unsigned long long[256]
#define OFF_CNT     2048     // float[256] per-object counts
#define OFF_OBJX    3072     // float[256] x_alpha
#define OFF_OBJY    4096     // float[256] y_alpha
#define OFF_OBJR2   5120     // float[256] |x_alpha|^2
#define OFF_OBJQA   6144     // float[256] q_alpha
#define OFF_OBJBA   7168     // float[256] beta_alpha
#define OFF_ATT     8192     // float[256] attractive sums
#define OFF_REP     9216     // float[256] repulsive sums
#define OFF_PDEN    10240    // float[256] payload denominators
#define OFF_PNUM    11264    // float[1024] payload numerators (4 per object)
#define OFF_SCAL    15360    // float[2]: noise beta sum, noise count
#define WS_BYTES    15368

__device__ __forceinline__ float clip_beta(float pb) {
    return fminf(fmaxf(pb, 0.0f), BETA_CLIP_C);
}
__device__ __forceinline__ float qfun(float pb) {
    float b = clip_beta(pb);
    float at = atanhf(b);
    return at * at + Q_MIN_C;
}

// ---- K0: zero workspace ----
__global__ void k_init(unsigned int* ws, int nwords) {
    int i = blockIdx.x * blockDim.x + threadIdx.x;
    if (i < nwords) ws[i] = 0u;
}

// ---- K1: per-point stats: counts, argmax(beta) per object, noise sums ----
__global__ void k_point_stats(const float* __restrict__ pred_beta,
                              const int* __restrict__ t_idx,
                              unsigned long long* __restrict__ amax,
                              float* __restrict__ cnt,
                              float* __restrict__ noise, int N) {
    int i = blockIdx.x * blockDim.x + threadIdx.x;
    if (i >= N) return;
    float b = clip_beta(pred_beta[i]);
    int t = t_idx[i];
    if (t >= 0 && t < KOBJ) {
        atomicAdd(&cnt[t], 1.0f);
        // beta in (0,1): positive-float bit pattern is order-preserving.
        // tie-break: smaller i wins => larger (0x7FFFFFFF - i)
        unsigned long long key =
            (((unsigned long long)__float_as_uint(b)) << 32) |
            (unsigned long long)(0x7FFFFFFFu - (unsigned)i);
        atomicMax(&amax[t], key);
    } else {
        atomicAdd(&noise[0], b);
        atomicAdd(&noise[1], 1.0f);
    }
}

// ---- K2: gather alpha-point data per object ----
__global__ void k_objects(const float* __restrict__ pred_beta,
                          const float* __restrict__ cc,
                          const unsigned long long* __restrict__ amax,
                          const float* __restrict__ cnt,
                          float* __restrict__ ox, float* __restrict__ oy,
                          float* __restrict__ or2, float* __restrict__ oqa,
                          float* __restrict__ oba) {
    int k = threadIdx.x;
    float x = 0.f, y = 0.f, r2 = 0.f, qa = 0.f, ba = 0.f;
    if (cnt[k] > 0.f) {
        unsigned long long key = amax[k];
        int alpha = (int)(0x7FFFFFFFu - (unsigned)(key & 0xFFFFFFFFull));
        x = cc[2 * alpha];
        y = cc[2 * alpha + 1];
        ba = clip_beta(pred_beta[alpha]);
        float at = atanhf(ba);
        qa = at * at + Q_MIN_C;
        r2 = x * x + y * y;
    }
    ox[k] = x; oy[k] = y; or2[k] = r2; oqa[k] = qa; oba[k] = ba;
}

// ---- K3: fused N x K pairwise pass via V_WMMA_F32_16X16X4_F32 ----
// Block = 256 threads = 8 waves; each wave owns 16 rows (points), sweeps the
// 16 column tiles (256 objects). A = 16x4 point coords (K=2..3 zero-padded),
// B = 4x16 object coords, D = 16x16 dot products. D layout: lane owns one
// column n = tile*16 + (lane&15); d[r] = rows 0..7 (lanes<16) / 8..15.
// Epilogue is fully branchless so EXEC stays all-ones around every WMMA.
__global__ __launch_bounds__(256)
void k_pairs(const float* __restrict__ cc,
             const float* __restrict__ pred_beta,
             const int* __restrict__ t_idx,
             const float* __restrict__ ox, const float* __restrict__ oy,
             const float* __restrict__ or2, const float* __restrict__ oqa,
             float* __restrict__ g_att, float* __restrict__ g_rep, int N) {
    __shared__ float s_ox[KOBJ], s_oy[KOBJ], s_or2[KOBJ], s_oqa[KOBJ];
    __shared__ float s_att[KOBJ], s_rep[KOBJ];
    int tid = threadIdx.x;
    s_ox[tid] = ox[tid]; s_oy[tid] = oy[tid];
    s_or2[tid] = or2[tid]; s_oqa[tid] = oqa[tid];
    s_att[tid] = 0.f; s_rep[tid] = 0.f;
    __syncthreads();

    int wave = tid >> 5;
    int lane = tid & 31;
    int rbase = blockIdx.x * 128 + wave * 16;
    bool lo = lane < 16;

    // A operand: lanes 0-15 hold row (rbase+lane) coords in (v0,v1)=(K0,K1);
    // lanes 16-31 hold K2,K3 = zero padding.
    v2f a = {0.f, 0.f};
    {
        int m = rbase + (lane & 15);
        float axv = 0.f, ayv = 0.f;
        if (m < N) { axv = cc[2 * m]; ayv = cc[2 * m + 1]; }
        a.x = lo ? axv : 0.f;
        a.y = lo ? ayv : 0.f;
    }
    // Per-lane row data for the 8 D-rows this lane owns.
    float qrow[8], r2row[8];
    int trow[8];
#pragma unroll
    for (int r = 0; r < 8; ++r) {
        int m = rbase + (lo ? r : 8 + r);
        if (m < N) {
            qrow[r] = qfun(pred_beta[m]);
            float cx = cc[2 * m], cyv = cc[2 * m + 1];
            r2row[r] = cx * cx + cyv * cyv;
            trow[r] = t_idx[m];
        } else {
            qrow[r] = 0.f; r2row[r] = 0.f; trow[r] = -2; // q=0 kills contribution
        }
    }

    v8f czero = {};
#pragma unroll 2
    for (int tile = 0; tile < 16; ++tile) {
        int n = tile * 16 + (lane & 15);
        // B operand: unconditional LDS loads, zero hi-half lanes via select.
        float bx = s_ox[n], by = s_oy[n];
        v2f b;
        b.x = lo ? bx : 0.f;
        b.y = lo ? by : 0.f;
        float r2o = s_or2[n];
        float qa  = s_oqa[n];

        v8f d = __builtin_amdgcn_wmma_f32_16x16x4_f32(
            false, a, false, b, (short)0, czero, false, false);

        float attacc = 0.f, repacc = 0.f;
#pragma unroll
        for (int r = 0; r < 8; ++r) {
            float distsq = fmaf(-2.f, d[r], r2row[r] + r2o);
            float dd = __builtin_amdgcn_sqrtf(distsq + 1e-6f); // v_sqrt_f32
            float h = fmaxf(1.f - dd, 0.f);                    // hinge
            bool mem = (trow[r] == n);
            float qm = mem ? qrow[r] : 0.f;   // attractive weight (member)
            float qn = mem ? 0.f : qrow[r];   // repulsive weight (non-member)
            attacc = fmaf(distsq, qm, attacc);
            repacc = fmaf(h, qn, repacc);
        }
        atomicAdd(&s_att[n], attacc * qa);
        atomicAdd(&s_rep[n], repacc * qa);
    }
    __syncthreads();
    atomicAdd(&g_att[tid], s_att[tid]);
    atomicAdd(&g_rep[tid], s_rep[tid]);
}

// ---- K4: payload terms per point ----
__global__ void k_payload(const float* __restrict__ pred_beta,
                          const float* __restrict__ pred_energy,
                          const float* __restrict__ pred_pos,
                          const float* __restrict__ pred_time,
                          const float* __restrict__ pred_id,
                          const float* __restrict__ t_energy,
                          const float* __restrict__ t_pos,
                          const float* __restrict__ t_time,
                          const int* __restrict__ t_idx,
                          float* __restrict__ pnum, float* __restrict__ pden,
                          int N) {
    int i = blockIdx.x * blockDim.x + threadIdx.x;
    if (i >= N) return;
    int t = t_idx[i];
    if (t < 0 || t >= KOBJ) return;
    float pb = pred_beta[i];
    float pw = clip_beta(pb);            // beta * not_noise (not_noise==1 here)
    float tE = t_energy[i];
    float ew = (tE > 10.f) ? 1.f : (tE - 0.5f) / 9.5f;
    ew = fmaxf(ew, 0.f);
    float de = tE - pred_energy[i];
    float denE = tE + 1.0f;                                  // ENERGY_DEN_OFFSET
    float e_loss = (denE == 0.f) ? 0.f : (de * de) / denE;   // divide_no_nan
    float dx = t_pos[2 * i] - pred_pos[2 * i];
    float dy = t_pos[2 * i + 1] - pred_pos[2 * i + 1];
    float p_loss = (dx * dx + dy * dy) * 0.01f;
    float dt = t_time[i] - pred_time[i];
    float tm_loss = dt * dt;
    float s = 0.f;
#pragma unroll
    for (int j = 0; j < 6; ++j) { float v = pred_id[6 * i + j]; s += v * v; }
    float cls_loss = 1e-8f * (s * (1.f / 6.f));
    float p0 = e_loss * ew, p1 = p_loss * ew, p2 = tm_loss * ew, p3 = cls_loss * ew;
    if (pb < 0.1f) { p0 = p1 = p2 = p3 = 0.f; }              // PAYLOAD_BETA_CLIP
    atomicAdd(&pnum[4 * t + 0], p0 * pw);
    atomicAdd(&pnum[4 * t + 1], p1 * pw);
    atomicAdd(&pnum[4 * t + 2], p2 * pw);
    atomicAdd(&pnum[4 * t + 3], p3 * pw);
    atomicAdd(&pden[t], pw);
}

// ---- K5: final reduction over objects -> scalar loss ----
__global__ void k_final(const float* __restrict__ cnt,
                        const float* __restrict__ g_att,
                        const float* __restrict__ g_rep,
                        const float* __restrict__ oba,
                        const float* __restrict__ pnum,
                        const float* __restrict__ pden,
                        const float* __restrict__ noise,
                        float* __restrict__ out, int N) {
    __shared__ float s1[KOBJ], s2[KOBJ];
    int k = threadIdx.x;
    float c = cnt[k];
    float valid = (c > 0.f) ? 1.f : 0.f;
    float local = 0.f;
    if (valid > 0.f) {
        local += g_att[k] / fmaxf(c, 1.f);                       // V_att_k
        local += g_rep[k] / fmaxf((float)N - c, 1.f);            // V_rep_k
        local += 1.f - oba[k];                                   // L_beta_k
        float num = pnum[4 * k] + pnum[4 * k + 1] + pnum[4 * k + 2] + pnum[4 * k + 3];
        local += num / fmaxf(pden[k], 1e-6f);                    // L_payload_k
    }
    s1[k] = local; s2[k] = valid;
    __syncthreads();
    for (int s = 128; s > 0; s >>= 1) {
        if (k < s) { s1[k] += s1[k + s]; s2[k] += s2[k + s]; }
        __syncthreads();
    }
    if (k == 0) {
        float n_valid = fmaxf(s2[0], 1.f);
        float L_noise = noise[0] / fmaxf(noise[1], 1.f);         // S_B = 1.0
        out[0] = s1[0] / n_valid + L_noise;
    }
}

extern "C" void kernel_launch(void* const* d_in, const int* in_sizes, int n_in,
                              void* d_out, int out_size, void* d_ws, size_t ws_size,
                              hipStream_t stream) {
    const float* pred_beta   = (const float*)d_in[0];
    const float* pred_cc     = (const float*)d_in[1];
    const float* pred_energy = (const float*)d_in[2];
    const float* pred_pos    = (const float*)d_in[3];
    const float* pred_time   = (const float*)d_in[4];
    const float* pred_id     = (const float*)d_in[5];
    const float* t_energy    = (const float*)d_in[6];
    const float* t_pos       = (const float*)d_in[7];
    const float* t_time      = (const float*)d_in[8];
    // d_in[9]  = t_pid  (unused by the loss)
    const int*   t_idx       = (const int*)d_in[10];
    // d_in[11] = rowsplits (single segment -> unused)
    int N = in_sizes[0];

    char* ws = (char*)d_ws;
    unsigned long long* amax = (unsigned long long*)(ws + OFF_AMAX);
    float* cnt   = (float*)(ws + OFF_CNT);
    float* ox    = (float*)(ws + OFF_OBJX);
    float* oy    = (float*)(ws + OFF_OBJY);
    float* or2   = (float*)(ws + OFF_OBJR2);
    float* oqa   = (float*)(ws + OFF_OBJQA);
    float* oba   = (float*)(ws + OFF_OBJBA);
    float* g_att = (float*)(ws + OFF_ATT);
    float* g_rep = (float*)(ws + OFF_REP);
    float* pden  = (float*)(ws + OFF_PDEN);
    float* pnum  = (float*)(ws + OFF_PNUM);
    float* noise = (float*)(ws + OFF_SCAL);

    int nwords = WS_BYTES / 4;
    k_init<<<(nwords + 255) / 256, 256, 0, stream>>>((unsigned int*)ws, nwords);
    k_point_stats<<<(N + 255) / 256, 256, 0, stream>>>(pred_beta, t_idx, amax, cnt, noise, N);
    k_objects<<<1, 256, 0, stream>>>(pred_beta, pred_cc, amax, cnt, ox, oy, or2, oqa, oba);
    k_pairs<<<(N + 127) / 128, 256, 0, stream>>>(pred_cc, pred_beta, t_idx,
                                                 ox, oy, or2, oqa, g_att, g_rep, N);
    k_payload<<<(N + 255) / 256, 256, 0, stream>>>(pred_beta, pred_energy, pred_pos,
                                                   pred_time, pred_id, t_energy, t_pos,
                                                   t_time, t_idx, pnum, pden, N);
    k_final<<<1, 256, 0, stream>>>(cnt, g_att, g_rep, oba, pnum, pden, noise,
                                   (float*)d_out, N);
}